// EMAVectorQuantizer_47605417509302
// MI455X (gfx1250) — compile-verified
//
#include <hip/hip_runtime.h>
#include <stdint.h>

typedef __attribute__((ext_vector_type(2))) float v2f;
typedef __attribute__((ext_vector_type(4))) float f4;
typedef __attribute__((ext_vector_type(8))) float v8f;

#define BATCH   16
#define DDIM    256
#define HH      32
#define WW      32
#define NTOT    16384          // B*H*W
#define KCODES  2048

#define ROWS    64             // z rows per workgroup (4 waves x 16)
#define CTILE   32             // codebook rows per K-step
#define NSTEPS  (KCODES / CTILE)
#define ZS      260            // padded LDS row stride (floats): conflict-free b64 frag loads, 16B-aligned rows
#define CS      260

// ---------------------------------------------------------------------------
// Kernel 1: codebook squared norms ||c_k||^2  (one wave per code row)
// ---------------------------------------------------------------------------
__global__ void vq_cnorm(const float* __restrict__ cb, float* __restrict__ cn) {
    const int wave = threadIdx.x >> 5;
    const int lane = threadIdx.x & 31;
    const int row  = blockIdx.x * 8 + wave;           // 256 threads = 8 waves
    const float* r = cb + (size_t)row * DDIM;
    float s = 0.f;
#pragma unroll
    for (int i = 0; i < DDIM; i += 32) {
        float v = r[i + lane];
        s = fmaf(v, v, s);
    }
#pragma unroll
    for (int m = 16; m >= 1; m >>= 1) s += __shfl_xor(s, m, 32);
    if (lane == 0) cn[row] = s;
}

// ---------------------------------------------------------------------------
// Kernel 2: fused GEMM (z . c^T via v_wmma_f32_16x16x4_f32) + running argmin.
// Block: 128 threads (4 waves). Each wave owns 16 rows, computes 16x32 strips.
// Codebook tiles are double-buffered in LDS and streamed with
// GLOBAL_LOAD_ASYNC_TO_LDS_B128 (ASYNCcnt), overlapping DMA with WMMA.
// ---------------------------------------------------------------------------
__global__ void __launch_bounds__(128)
vq_argmin(const float* __restrict__ z, const float* __restrict__ cb,
          const float* __restrict__ cn, int* __restrict__ idx_out) {
    extern __shared__ float smem[];
    float* zt  = smem;                       // ROWS x ZS        (z tile)
    float* cbt = zt + ROWS * ZS;             // 2 x CTILE x CS   (codebook ping-pong)
    float* cns = cbt + 2 * CTILE * CS;       // KCODES           (code norms)

    const int tid   = threadIdx.x;
    const int lane  = tid & 31;
    const int wave  = tid >> 5;
    const int laneN = lane & 15;
    const int hi    = lane >> 4;             // 0: K(+0,+1)  1: K(+2,+3)

    const int n0  = blockIdx.x * ROWS;       // 64-row tile; H*W=1024 => b fixed
    const int b   = n0 >> 10;
    const int hw0 = n0 & 1023;
    const float* zg = z + (size_t)b * (DDIM * HH * WW) + hw0;

    // LDS byte offset of the codebook ping-pong region (flat->LDS aperture:
    // low 32 bits of a shared-memory flat address are the LDS offset).
    const uint32_t cbt_lds = (uint32_t)(uintptr_t)cbt;

    // Issue one 32KB tile as 16 async b128 copies per thread (no VGPR staging).
    auto issue_tile = [&](int colbase, int buf) {
#pragma unroll
        for (int j = 0; j < 16; ++j) {
            const int x  = tid + 128 * j;
            const int dq = x & 63;                       // float4 index in row
            const int c  = x >> 6;                       // code row in tile
            const float*   g = cb + (size_t)(colbase + c) * DDIM + dq * 4;
            const uint32_t l = cbt_lds +
                (uint32_t)(((buf * CTILE + c) * CS + dq * 4) * sizeof(float));
            asm volatile("global_load_async_to_lds_b128 %0, %1, off"
                         :: "v"(l), "v"(g) : "memory");
        }
    };

    // Prologue: kick off tile 0 DMA, then fill cnorm + z tile while it flies.
    issue_tile(0, 0);

    for (int i = tid; i < KCODES; i += 128) cns[i] = cn[i];

    // z tile -> LDS. z layout (B,D,H,W): row n, col d lives at d*1024 + (n&1023).
    // float4 along n (contiguous w), scalar scatter into padded LDS rows.
    for (int x = tid; x < 16 * DDIM; x += 128) {
        const int i4 = x & 15;               // n-subgroup (4 rows)
        const int d  = x >> 4;
        f4 v = *reinterpret_cast<const f4*>(zg + (size_t)d * (HH * WW) + i4 * 4);
        float* dst = zt + (i4 * 4) * ZS + d;
        dst[0 * ZS] = v.x; dst[1 * ZS] = v.y; dst[2 * ZS] = v.z; dst[3 * ZS] = v.w;
    }

    asm volatile("s_wait_asynccnt 0" ::: "memory");
    __syncthreads();

    // Fragment base pointers per ISA VGPR layouts:
    //  A 16x4 f32: lanes 0-15 & 16-31 both hold M=laneN; hi selects K+{0,1}/{2,3}
    //  B 4x16 f32: lane holds column N=laneN; hi selects K rows the same way
    const float* ztw = zt + (wave * 16 + laneN) * ZS + hi * 2;

    float minv[8];
    int   mini[8];
#pragma unroll
    for (int v = 0; v < 8; ++v) { minv[v] = 3.402823466e+38f; mini[v] = 0; }

    for (int step = 0; step < NSTEPS; ++step) {
        const int cur     = step & 1;
        const int colbase = step * CTILE;

        // Start DMA for the next tile into the inactive buffer.
        if (step + 1 < NSTEPS) issue_tile(colbase + CTILE, cur ^ 1);

        const float* bt0 = cbt + cur * (CTILE * CS) + laneN * CS + hi * 2;
        const float* bt1 = bt0 + 16 * CS;

        v8f acc0 = {};
        v8f acc1 = {};
#pragma unroll 8
        for (int kk = 0; kk < DDIM / 4; ++kk) {
            v2f a  = *reinterpret_cast<const v2f*>(ztw + kk * 4);
            v2f b0 = *reinterpret_cast<const v2f*>(bt0 + kk * 4);
            v2f b1 = *reinterpret_cast<const v2f*>(bt1 + kk * 4);
            acc0 = __builtin_amdgcn_wmma_f32_16x16x4_f32(
                       false, a, false, b0, (short)0, acc0, false, false);
            acc1 = __builtin_amdgcn_wmma_f32_16x16x4_f32(
                       false, a, false, b1, (short)0, acc1, false, false);
        }

        // epilogue: dist = ||c||^2 - 2*dot ; fold into running per-lane min
        const float cn0 = cns[colbase + laneN];
        const float cn1 = cns[colbase + 16 + laneN];
        const int   i0  = colbase + laneN;
        const int   i1  = i0 + 16;
#pragma unroll
        for (int v = 0; v < 8; ++v) {
            float c0 = fmaf(-2.f, acc0[v], cn0);
            float c1 = fmaf(-2.f, acc1[v], cn1);
            bool  s1 = (c1 < c0) || (c1 == c0 && i1 < i0);
            float cm = s1 ? c1 : c0;
            int   ci = s1 ? i1 : i0;
            bool  lt = (cm < minv[v]) || (cm == minv[v] && ci < mini[v]);
            minv[v] = lt ? cm : minv[v];
            mini[v] = lt ? ci : mini[v];
        }

        // This wave's next-tile DMA done; barrier makes it visible to all waves
        // and guarantees everyone finished reading the current buffer.
        asm volatile("s_wait_asynccnt 0" ::: "memory");
        __syncthreads();
    }

    // cross-lane argmin within each 16-lane half (C/D layout: lanes<16 row v,
    // lanes>=16 row v+8), keeping lowest index on ties
#pragma unroll
    for (int v = 0; v < 8; ++v) {
        float m  = minv[v];
        int   ix = mini[v];
#pragma unroll
        for (int s = 8; s >= 1; s >>= 1) {
            float om = __shfl_xor(m, s, 32);
            int   oi = __shfl_xor(ix, s, 32);
            bool take = (om < m) || (om == m && oi < ix);
            m  = take ? om : m;
            ix = take ? oi : ix;
        }
        if (laneN == 0)
            idx_out[n0 + wave * 16 + hi * 8 + v] = ix;
    }
}

// ---------------------------------------------------------------------------
// Kernel 3: gather + transpose: out[b][d][h][w] = codebook[idx[n]][d]
// One block per (b,h): stage 32 selected rows in LDS (coalesced reads),
// then write w-contiguous (fully coalesced) stores.
// ---------------------------------------------------------------------------
__global__ void vq_gather(const float* __restrict__ cb,
                          const int* __restrict__ idx,
                          float* __restrict__ out) {
    __shared__ int   sidx[WW];
    __shared__ float tile[WW][CS];           // 32 x 260 floats, 16B-aligned rows

    const int h   = blockIdx.x;
    const int b   = blockIdx.y;
    const int tid = threadIdx.x;

    if (tid < WW) sidx[tid] = idx[(b * HH + h) * WW + tid];
    __syncthreads();

    for (int x = tid; x < WW * (DDIM / 4); x += 256) {
        const int dq = x & 63;
        const int w  = x >> 6;
        f4 v = *reinterpret_cast<const f4*>(cb + (size_t)sidx[w] * DDIM + dq * 4);
        *reinterpret_cast<f4*>(&tile[w][dq * 4]) = v;
    }
    __syncthreads();

    const int w     = tid & 31;
    const int dslot = tid >> 5;              // 0..7
    float* og = out + ((size_t)b * DDIM * HH + h) * WW + w;
#pragma unroll 4
    for (int j = 0; j < 32; ++j) {
        const int d = dslot * 32 + j;
        og[(size_t)d * (HH * WW)] = tile[w][d];
    }
}

// ---------------------------------------------------------------------------
extern "C" void kernel_launch(void* const* d_in, const int* in_sizes, int n_in,
                              void* d_out, int out_size, void* d_ws, size_t ws_size,
                              hipStream_t stream) {
    (void)in_sizes; (void)n_in; (void)out_size; (void)ws_size;
    const float* z  = (const float*)d_in[0];
    const float* cb = (const float*)d_in[1];
    float* cn  = (float*)d_ws;                                   // 2048 f32
    int*   idx = (int*)((char*)d_ws + KCODES * sizeof(float));   // 16384 i32
    float* out = (float*)d_out;

    vq_cnorm<<<KCODES / 8, 256, 0, stream>>>(cb, cn);

    const size_t smem = (size_t)(ROWS * ZS + 2 * CTILE * CS + KCODES) * sizeof(float);
    vq_argmin<<<NTOT / ROWS, 128, smem, stream>>>(z, cb, cn, idx);

    dim3 g(HH, BATCH);
    vq_gather<<<g, 256, 0, stream>>>(cb, idx, out);
}